// Model_21766894256250
// MI455X (gfx1250) — compile-verified
//
#include <hip/hip_runtime.h>
#include <stdint.h>

// ---------------------------------------------------------------------------
// SHA-256 single-block compression on MI455X (gfx1250).
//
// Latency-bound serial 64-round chain -> single wave32, all state in VGPRs.
// WMMA/TDM deliberately unused: dot-product units cannot express mod-2^32
// carries / 32-bit rotates and would only add latency to the chain.
//
// Key trick this revision: keep the ENTIRE round chain on VALU with zero
// SALU<->VALU crossings.  `asm("" : "+v"(x))` emits no instruction but pins
// a value to a VGPR and makes it look divergent, so the compiler lowers the
// whole chain with fused VALU ops it pattern-matches itself:
//     rotr -> v_alignbit_b32      (via __builtin_rotateright32 / fshr)
//     ch   -> v_bfi_b32
//     maj  -> v_and_or_b32 / v_bfi_b32
//     ^^   -> v_xor3_b32,  ++ -> v_add3_u32
// No inline-asm compute blocks => no scheduler fences => dual-issue pairing
// and s_delay_alu placement work.  K stays in SGPRs (free VALU scalar src).
// ---------------------------------------------------------------------------

__device__ __forceinline__ uint32_t vgpr(uint32_t x) {
  asm("" : "+v"(x));   // zero-cost: force VGPR residency + divergent-looking
  return x;
}

__device__ __forceinline__ uint32_t rotr(uint32_t x, int n) {
  return __builtin_rotateright32(x, n);        // -> single v_alignbit_b32
}
__device__ __forceinline__ uint32_t ch_f(uint32_t x, uint32_t y, uint32_t z) {
  return z ^ (x & (y ^ z));                    // -> v_bfi_b32
}
__device__ __forceinline__ uint32_t maj_f(uint32_t x, uint32_t y, uint32_t z) {
  return (x & y) | (z & (x | y));              // -> v_and_or_b32 (+v_and)
}
__device__ __forceinline__ uint32_t Sig0(uint32_t x) {
  return rotr(x, 2) ^ rotr(x, 13) ^ rotr(x, 22);   // 3x alignbit + v_xor3
}
__device__ __forceinline__ uint32_t Sig1(uint32_t x) {
  return rotr(x, 6) ^ rotr(x, 11) ^ rotr(x, 25);
}
__device__ __forceinline__ uint32_t Gam0(uint32_t x) {
  return rotr(x, 7) ^ rotr(x, 18) ^ (x >> 3);
}
__device__ __forceinline__ uint32_t Gam1(uint32_t x) {
  return rotr(x, 17) ^ rotr(x, 19) ^ (x >> 10);
}

__global__ void __launch_bounds__(32)
sha256_single_block(const uint32_t* __restrict__ msg,   // 64 byte-values
                    const uint32_t* __restrict__ K,     // 64 round constants
                    const uint32_t* __restrict__ H0,    // 8 IV words
                    uint32_t* __restrict__ out)         // 8 digest words
{
  // --- pack 64 bytes into 16 big-endian words on SALU (overlaps SMEM
  //     latency), then launder each word into a VGPR at zero cost. ---
  uint32_t W[16];
#pragma unroll
  for (int i = 0; i < 16; ++i) {
    uint32_t w = (msg[4 * i + 0] << 24) | (msg[4 * i + 1] << 16) |
                 (msg[4 * i + 2] << 8)  |  msg[4 * i + 3];
    W[i] = vgpr(w);
  }

  uint32_t a = vgpr(H0[0]), b = vgpr(H0[1]), c = vgpr(H0[2]), d = vgpr(H0[3]);
  uint32_t e = vgpr(H0[4]), f = vgpr(H0[5]), g = vgpr(H0[6]), h = vgpr(H0[7]);

  // --- 64 fully-unrolled VALU rounds; W is a 16-entry register ring ---
#pragma unroll
  for (int i = 0; i < 64; ++i) {
    uint32_t wi;
    if (i < 16) {
      wi = W[i];
    } else {
      wi = Gam1(W[(i - 2) & 15]) + W[(i - 7) & 15] +
           Gam0(W[(i - 15) & 15]) + W[i & 15];
      W[i & 15] = wi;
    }
    // h + K[i] + wi has no dependence on rounds i-2..i (h == e three rounds
    // back): the scheduler hoists it off the critical e/a recurrences.
    uint32_t T1 = (h + K[i] + wi) + (Sig1(e) + ch_f(e, f, g));
    uint32_t T2 = Sig0(a) + maj_f(a, b, c);
    h = g; g = f; f = e;
    e = d + T1;
    d = c; c = b; b = a;
    a = T1 + T2;
  }

  if (threadIdx.x == 0) {
    uint4 lo = make_uint4(a + H0[0], b + H0[1], c + H0[2], d + H0[3]);
    uint4 hi = make_uint4(e + H0[4], f + H0[5], g + H0[6], h + H0[7]);
    *reinterpret_cast<uint4*>(out + 0) = lo;   // global_store_b128
    *reinterpret_cast<uint4*>(out + 4) = hi;   // global_store_b128
  }
}

extern "C" void kernel_launch(void* const* d_in, const int* in_sizes, int n_in,
                              void* d_out, int out_size, void* d_ws, size_t ws_size,
                              hipStream_t stream) {
  (void)in_sizes; (void)n_in; (void)out_size; (void)d_ws; (void)ws_size;
  const uint32_t* msg = (const uint32_t*)d_in[0];  // message: 64 byte values
  const uint32_t* K   = (const uint32_t*)d_in[1];  // 64 round constants
  const uint32_t* H0  = (const uint32_t*)d_in[2];  // 8 IV words
  uint32_t* out = (uint32_t*)d_out;                // 8 digest words

  // Single wave32: the problem is one serial dependency chain.
  sha256_single_block<<<1, 32, 0, stream>>>(msg, K, H0, out);
}